// MoE_53077205844425
// MI455X (gfx1250) — compile-verified
//
#include <hip/hip_runtime.h>
#include <hip/hip_bf16.h>

typedef __attribute__((ext_vector_type(2))) float v2f;
typedef __attribute__((ext_vector_type(4))) float v4f;
typedef __attribute__((ext_vector_type(8))) float v8f;

#define T_TOK 16384
#define H_DIM 1024
#define M_DIM 2048
#define E_NUM 16
#define TM 32
#define TN 256
#define MC 64
#define XPAD 4   // row stride H_DIM+4 floats: 16B-aligned rows, conflict-free cols
#define HPAD 4

// ---------------------------------------------------------------------------
// Gate: logits -> top2 -> renormalized weights -> per-expert routing lists
// One wave (32 lanes) per token.
// ---------------------------------------------------------------------------
__global__ __launch_bounds__(256)
void moe_gate_route(const float* __restrict__ X, const float* __restrict__ Wg,
                    const float* __restrict__ bg,
                    int* __restrict__ counts, int* __restrict__ tok_id,
                    float* __restrict__ tok_w) {
    const int wave = threadIdx.x >> 5;
    const int lane = threadIdx.x & 31;
    const long t = (long)blockIdx.x * 8 + wave;
    if (t >= T_TOK) return;

    const float* xrow = X + t * H_DIM;
    float acc[E_NUM];
#pragma unroll
    for (int e = 0; e < E_NUM; ++e) acc[e] = 0.f;

    for (int h = lane; h < H_DIM; h += 32) {
        const float x = xrow[h];
        const float* wrow = Wg + (long)h * E_NUM;
#pragma unroll
        for (int e = 0; e < E_NUM; ++e) acc[e] += x * wrow[e];
    }
#pragma unroll
    for (int e = 0; e < E_NUM; ++e) {
#pragma unroll
        for (int off = 16; off > 0; off >>= 1)
            acc[e] += __shfl_xor(acc[e], off, 32);
    }

    if (lane == 0) {
        float lg[E_NUM];
#pragma unroll
        for (int e = 0; e < E_NUM; ++e) lg[e] = acc[e] + bg[e];
        float b0 = lg[0]; int i0 = 0;
#pragma unroll
        for (int e = 1; e < E_NUM; ++e) if (lg[e] > b0) { b0 = lg[e]; i0 = e; }
        float b1v = -3.4e38f; int i1 = 0;
#pragma unroll
        for (int e = 0; e < E_NUM; ++e)
            if (e != i0 && lg[e] > b1v) { b1v = lg[e]; i1 = e; }
        const float w0 = 1.f / (1.f + expf(b1v - b0));
        const float w1 = 1.f - w0;

        int p0 = atomicAdd(&counts[i0], 1);
        tok_id[i0 * T_TOK + p0] = (int)t;
        tok_w [i0 * T_TOK + p0] = w0;
        int p1 = atomicAdd(&counts[i1], 1);
        tok_id[i1 * T_TOK + p1] = (int)t;
        tok_w [i1 * T_TOK + p1] = w1;
    }
}

// ---------------------------------------------------------------------------
// Expert FFN: out[t] += w * ( relu(x @ W1 + b1) @ W2 + b2 )
// Block = 32 gathered tokens x 256 output cols, 8 waves, f32 WMMA 16x16x4.
// ---------------------------------------------------------------------------
__global__ __launch_bounds__(256)
void moe_expert_ffn(const float* __restrict__ X,
                    const float* __restrict__ W1, const float* __restrict__ b1,
                    const float* __restrict__ W2, const float* __restrict__ b2,
                    const int* __restrict__ counts,
                    const int* __restrict__ tok_id,
                    const float* __restrict__ tok_w,
                    float* __restrict__ out) {
    __shared__ float Xs[TM][H_DIM + XPAD];
    __shared__ float Hs[2][TM][MC + HPAD];   // double-buffered h tile
    __shared__ int   s_tid[TM];
    __shared__ float s_tw[TM];

    const int e   = blockIdx.z;
    const int cnt = counts[e];
    const int row0 = blockIdx.x * TM;
    if (row0 >= cnt) return;
    const int col0 = blockIdx.y * TN;
    const int tid  = threadIdx.x;
    const int wave = tid >> 5;
    const int lane = tid & 31;
    const int rows = min(TM, cnt - row0);

    if (tid < TM) {
        if (tid < rows) {
            s_tid[tid] = tok_id[(long)e * T_TOK + row0 + tid];
            s_tw [tid] = tok_w [(long)e * T_TOK + row0 + tid];
        } else {
            s_tid[tid] = 0;
            s_tw [tid] = 0.f;
        }
    }
    __syncthreads();

    // gather 32 token rows of X into LDS, 16B vectors (coalesced b128)
    {
        const int HV = H_DIM / 4;             // 256 vec4 per row
        for (int idx = tid; idx < TM * HV; idx += 256) {
            const int r = idx >> 8;
            const int c = (idx & (HV - 1)) * 4;
            v4f v = {0.f, 0.f, 0.f, 0.f};
            if (r < rows) v = *(const v4f*)(X + (long)s_tid[r] * H_DIM + c);
            *(v4f*)&Xs[r][c] = v;
        }
    }
    __syncthreads();

    const int rb    = wave & 1;           // token-row block (0..1)
    const int cb    = wave >> 1;          // col block (0..3)
    const int halfk = lane >> 4;          // 0 or 1
    const int ln    = lane & 15;

    const float* W1e = W1 + (long)e * H_DIM * M_DIM;
    const float* W2e = W2 + (long)e * M_DIM * H_DIM;

    v8f acc[4];
#pragma unroll
    for (int j = 0; j < 4; ++j)
#pragma unroll
        for (int i = 0; i < 8; ++i) acc[j][i] = 0.f;

    int buf = 0;
    for (int mc = 0; mc < M_DIM; mc += MC) {
        // ---- fc1: Hs[buf] = relu(Xs @ W1[:, mc+cb*16 .. +16] + b1) ---------
        // 4 rotating accumulators -> WMMA RAW chain distance 4
        v8f c1[4];
        {
            const float bias = b1[(long)e * M_DIM + mc + cb * 16 + ln];
#pragma unroll
            for (int i = 0; i < 8; ++i) {
                c1[0][i] = bias;
                c1[1][i] = 0.f; c1[2][i] = 0.f; c1[3][i] = 0.f;
            }
        }
        const float* w1col = W1e + (mc + cb * 16 + ln);
        const float* xrow  = &Xs[rb * 16 + ln][0];
        for (int k = 0; k < H_DIM; k += 16) {
#pragma unroll
            for (int u = 0; u < 4; ++u) {
                const int kk = k + 4 * u + 2 * halfk;
                v2f a = *(const v2f*)(xrow + kk);
                v2f b;
                b.x = w1col[(long)kk * M_DIM];
                b.y = w1col[(long)(kk + 1) * M_DIM];
                c1[u] = __builtin_amdgcn_wmma_f32_16x16x4_f32(
                            false, a, false, b, (short)0, c1[u], false, false);
            }
        }
#pragma unroll
        for (int i = 0; i < 8; ++i) {
            const float v = (c1[0][i] + c1[1][i]) + (c1[2][i] + c1[3][i]);
            Hs[buf][rb * 16 + i + 8 * halfk][cb * 16 + ln] = v > 0.f ? v : 0.f;
        }
        __syncthreads();   // one barrier per chunk (double-buffered Hs)

        // ---- fc2: acc += Hs[buf] @ W2[mc .. mc+MC, col0 + cb*64 + j*16] ----
        const float* hrow = &Hs[buf][rb * 16 + ln][0];
#pragma unroll 2
        for (int k = 0; k < MC; k += 4) {
            const int kk = k + 2 * halfk;
            v2f a = *(const v2f*)(hrow + kk);
            const float* w2r0 = W2e + (long)(mc + kk) * H_DIM + col0 + cb * 64 + ln;
            const float* w2r1 = w2r0 + H_DIM;
#pragma unroll
            for (int j = 0; j < 4; ++j) {
                v2f b;
                b.x = w2r0[j * 16];
                b.y = w2r1[j * 16];
                acc[j] = __builtin_amdgcn_wmma_f32_16x16x4_f32(
                            false, a, false, b, (short)0, acc[j], false, false);
            }
        }
        buf ^= 1;
    }

    // epilogue: out[t][col] += w * (acc + b2[col]); exactly 2 commutative
    // atomic adds per output element -> bitwise deterministic.
#pragma unroll
    for (int j = 0; j < 4; ++j) {
        const int colj = col0 + cb * 64 + j * 16 + ln;
        const float bias2 = b2[(long)e * H_DIM + colj];
#pragma unroll
        for (int i = 0; i < 8; ++i) {
            const int r = rb * 16 + i + 8 * halfk;
            if (r < rows) {
                const float val = s_tw[r] * (acc[j][i] + bias2);
                atomicAdd(&out[(long)s_tid[r] * H_DIM + colj], val);
            }
        }
    }
}

// ---------------------------------------------------------------------------
extern "C" void kernel_launch(void* const* d_in, const int* in_sizes, int n_in,
                              void* d_out, int out_size, void* d_ws, size_t ws_size,
                              hipStream_t stream) {
    const float* X  = (const float*)d_in[0];
    const float* Wg = (const float*)d_in[1];
    const float* bg = (const float*)d_in[2];
    const float* W1 = (const float*)d_in[3];
    const float* b1 = (const float*)d_in[4];
    const float* W2 = (const float*)d_in[5];
    const float* b2 = (const float*)d_in[6];
    float* out = (float*)d_out;

    // workspace layout: [counts 256B][tok_id E*T ints][tok_w E*T floats]
    int*   counts = (int*)d_ws;
    int*   tok_id = (int*)((char*)d_ws + 256);
    float* tok_w  = (float*)((char*)d_ws + 256 + sizeof(int) * (size_t)E_NUM * T_TOK);

    hipMemsetAsync(counts, 0, 256, stream);
    hipMemsetAsync(out, 0, sizeof(float) * (size_t)T_TOK * H_DIM, stream);

    moe_gate_route<<<T_TOK / 8, 256, 0, stream>>>(X, Wg, bg, counts, tok_id, tok_w);

    dim3 grid(T_TOK / TM, H_DIM / TN, E_NUM);
    moe_expert_ffn<<<grid, 256, 0, stream>>>(X, W1, b1, W2, b2,
                                             counts, tok_id, tok_w, out);
}